// Model_2241972928586
// MI455X (gfx1250) — compile-verified
//
#include <hip/hip_runtime.h>
#include <hip/hip_bf16.h>
#include <math.h>

#define N_NODES 50000
#define F_DIM 256
#define H_DIM 128
#define C_DIM 16
#define E_EDGES 800000
#define N_CAND 400000
#define CAND_PAD (1 << 19)                       // 524288 >= 400000
#define NNZ_TOT (E_EDGES + N_CAND + N_NODES)     // 1,250,000
#define N_LAYERS 2
#define ALPHA_C 0.5f
#define SRATE 0.5f
#define TAU_C 0.3f
#define EPS_C 1e-8f

typedef __attribute__((ext_vector_type(2))) float v2f;
typedef __attribute__((ext_vector_type(8))) float v8f;

// ---------------- helpers ----------------

__device__ __forceinline__ unsigned f_enc(float f) {   // order-preserving float->uint
  unsigned u = __float_as_uint(f);
  return (u & 0x80000000u) ? ~u : (u | 0x80000000u);
}
__device__ __forceinline__ float f_dec(unsigned u) {
  u = (u & 0x80000000u) ? (u & 0x7FFFFFFFu) : ~u;
  return __uint_as_float(u);
}

// threefry2x32 (5x4 rounds) — structural match of JAX's counter RNG
__device__ __forceinline__ void tfry(unsigned k0, unsigned k1, unsigned c0, unsigned c1,
                                     unsigned& o0, unsigned& o1) {
  const unsigned R[8] = {13u, 15u, 26u, 6u, 17u, 29u, 16u, 24u};
  unsigned ks0 = k0, ks1 = k1, ks2 = k0 ^ k1 ^ 0x1BD11BDAu;
  unsigned x0 = c0 + ks0, x1 = c1 + ks1;
#pragma unroll
  for (int r = 0; r < 5; ++r) {
#pragma unroll
    for (int i = 0; i < 4; ++i) {
      unsigned rot = R[((r & 1) << 2) | i];
      x0 += x1;
      x1 = (x1 << rot) | (x1 >> (32u - rot));
      x1 ^= x0;
    }
    unsigned a = ((r + 1) % 3 == 0) ? ks0 : (((r + 1) % 3 == 1) ? ks1 : ks2);
    unsigned b = ((r + 2) % 3 == 0) ? ks0 : (((r + 2) % 3 == 1) ? ks1 : ks2);
    x0 += a;
    x1 += b + (unsigned)(r + 1);
  }
  o0 = x0; o1 = x1;
}

__device__ __forceinline__ float u_clamped(unsigned bits) {
  // uniform in [1e-6, 1-1e-6)
  float u = (float)(bits >> 8) * (1.0f / 16777216.0f);
  return 1e-6f + u * (1.0f - 2e-6f);
}

// ---------------- kernel 1: h = LN(feature@W0+b0); energies = LSE(h@W1+b1) ----------------
// 256 threads = 8 waves; each wave one 16x16 tile of C via V_WMMA_F32_16X16X4_F32.

__global__ __launch_bounds__(256) void k_gemm0_ln_energy(
    const float* __restrict__ feat, const float* __restrict__ W0,
    const float* __restrict__ b0, const float* __restrict__ W1,
    const float* __restrict__ b1, const float* __restrict__ g,
    const float* __restrict__ bb, float* __restrict__ h_out,
    float* __restrict__ energies) {
  __shared__ float sm[16][H_DIM];
  __shared__ float rs[16][16], rq[16][16];
  __shared__ float rmean[16], rrstd[16];

  const int tid = threadIdx.x;
  const int wave = tid >> 5;
  const int lane = tid & 31;
  const int row0 = blockIdx.x * 16;
  const int c0 = wave * 16;
  const int hl = lane & 15;   // 0..15
  const int hi = lane >> 4;   // 0 or 1 (K sub-pair select per ISA f32 A/B layout)

  v8f acc = {};
  for (int k = 0; k < F_DIM; k += 4) {
    v2f a, b;
    const float* arow = feat + (size_t)(row0 + hl) * F_DIM + k + hi * 2;
    a[0] = arow[0];
    a[1] = arow[1];
    __builtin_prefetch(arow + 8, 0, 1);   // -> global_prefetch_b8
    const float* bk = W0 + (size_t)(k + hi * 2) * H_DIM + c0 + hl;
    b[0] = bk[0];
    b[1] = bk[H_DIM];
    acc = __builtin_amdgcn_wmma_f32_16x16x4_f32(false, a, false, b, (short)0, acc,
                                                false, false);
  }
  // C layout: VGPR r -> (M = r + 16hi?8:0, N = lane&15)
  const float bias = b0[c0 + hl];
#pragma unroll
  for (int r = 0; r < 8; ++r) sm[r + hi * 8][c0 + hl] = acc[r] + bias;
  __syncthreads();

  // LayerNorm over 128 cols: 16 threads/row, 8 cols each
  const int lrow = tid >> 4;
  const int sub = tid & 15;
  float s = 0.f, q = 0.f;
#pragma unroll
  for (int d = 0; d < 8; ++d) {
    float v = sm[lrow][sub * 8 + d];
    s += v;
    q += v * v;
  }
  rs[lrow][sub] = s;
  rq[lrow][sub] = q;
  __syncthreads();
  if (sub == 0) {
    float S = 0.f, Q = 0.f;
    for (int t = 0; t < 16; ++t) { S += rs[lrow][t]; Q += rq[lrow][t]; }
    float m = S * (1.0f / H_DIM);
    float var = Q * (1.0f / H_DIM) - m * m;
    rmean[lrow] = m;
    rrstd[lrow] = rsqrtf(var + 1e-5f);
  }
  __syncthreads();
#pragma unroll
  for (int d = 0; d < 8; ++d) {
    int col = sub * 8 + d;
    float y = (sm[lrow][col] - rmean[lrow]) * rrstd[lrow] * g[col] + bb[col];
    sm[lrow][col] = y;
    h_out[(size_t)(row0 + lrow) * H_DIM + col] = y;
  }
  __syncthreads();

  // energies = logsumexp(h@W1 + b1 + EPS): thread (lrow, c) computes one dot
  const int c = sub;
  float y = b1[c] + EPS_C;
  for (int k = 0; k < H_DIM; ++k) y += sm[lrow][k] * W1[k * C_DIM + c];
  rs[lrow][c] = y;
  __syncthreads();
  if (c == 0) {
    float mx = rs[lrow][0];
    for (int t = 1; t < 16; ++t) mx = fmaxf(mx, rs[lrow][t]);
    float se = 0.f;
    for (int t = 0; t < 16; ++t) se += expf(rs[lrow][t] - mx);
    energies[row0 + lrow] = mx + logf(se);
  }
}

// ---------------- candidate scoring: T = ||x[ni]-x[nj]|| * en[ni]*en[nj] ----------------
// one wave32 per candidate; 4 dims/lane; stats via block-reduced atomics

__global__ __launch_bounds__(256) void k_score(
    const float* __restrict__ x, const float* __restrict__ en,
    const int* __restrict__ cand, float* __restrict__ keyT, int* __restrict__ idxT,
    float* __restrict__ stats) {
  __shared__ float tloc[8];
  const int gw = (int)((blockIdx.x * blockDim.x + threadIdx.x) >> 5);  // all full blocks
  const int lane = threadIdx.x & 31;
  const int wv = threadIdx.x >> 5;
  const int ni = cand[gw];
  const int nj = cand[N_CAND + gw];
  const float* xi = x + (size_t)ni * H_DIM;
  const float* xj = x + (size_t)nj * H_DIM;
  float ss = 0.f;
#pragma unroll
  for (int d = 0; d < 4; ++d) {
    float dv = xi[lane * 4 + d] - xj[lane * 4 + d];
    ss += dv * dv;
  }
  for (int m = 16; m; m >>= 1) ss += __shfl_xor(ss, m, 32);
  float T = sqrtf(ss) * en[ni] * en[nj];
  if (lane == 0) {
    keyT[gw] = T;
    idxT[gw] = gw;
    tloc[wv] = T;
  }
  __syncthreads();
  if (threadIdx.x == 0) {
    float sum = 0.f, mn = tloc[0], mx = tloc[0];
    for (int t = 0; t < 8; ++t) {
      sum += tloc[t];
      mn = fminf(mn, tloc[t]);
      mx = fmaxf(mx, tloc[t]);
    }
    atomicAdd(&stats[0], sum);
    unsigned* su = (unsigned*)stats;
    atomicMin(&su[1], f_enc(mn));
    atomicMax(&su[2], f_enc(mx));
  }
}

__global__ void k_statinit(float* stats) {
  stats[0] = 0.f;
  unsigned* su = (unsigned*)stats;
  su[1] = 0xFFFFFFFFu;
  su[2] = 0u;
}

__global__ void k_pad(float* keyT, int* idxT) {
  int i = blockIdx.x * blockDim.x + threadIdx.x;
  if (i < CAND_PAD - N_CAND) {
    keyT[N_CAND + i] = -INFINITY;
    idxT[N_CAND + i] = N_CAND + i;
  }
}

// ---------------- bitonic sort stage: descending, stable via index tie-break ----------------

__device__ __forceinline__ bool precedes(float ka, int ia, float kb, int ib) {
  return (ka > kb) || (ka == kb && ia < ib);
}

__global__ __launch_bounds__(256) void k_bitonic(float* __restrict__ key,
                                                 int* __restrict__ idx, int j, int k) {
  unsigned i = blockIdx.x * blockDim.x + threadIdx.x;
  unsigned p = i ^ (unsigned)j;
  if (p <= i) return;
  float ka = key[i], kb = key[p];
  int ia = idx[i], ib = idx[p];
  bool desc = ((i & (unsigned)k) == 0);
  bool doswap = desc ? precedes(kb, ib, ka, ia) : precedes(ka, ia, kb, ib);
  if (doswap) {
    key[i] = kb; key[p] = ka;
    idx[i] = ib; idx[p] = ia;
  }
}

// ---------------- Gumbel-softmax edge selection ----------------

__global__ __launch_bounds__(256) void k_sel(
    const float* __restrict__ keyT, const int* __restrict__ idxT,
    const int* __restrict__ cand, const float* __restrict__ stats,
    int* __restrict__ rows, int* __restrict__ cols, int layer) {
  int j = blockIdx.x * blockDim.x + threadIdx.x;
  if (j >= N_CAND) return;
  const unsigned* su = (const unsigned*)stats;
  float Tmin = f_dec(su[1]);
  float Tmax = f_dec(su[2]);
  float range = Tmax - Tmin + EPS_C;
  float pmean = 1.f - (stats[0] * (1.0f / (float)N_CAND) - Tmin) / range;
  float T = keyT[j];  // already sorted keys = sT
  float p = 1.f - (T - Tmin) / range;
  p = fminf(p * (SRATE / (pmean + EPS_C)), 1.f);
  float l0 = logf(1.f - p + EPS_C);
  float l1 = logf(p + EPS_C);
  unsigned f0, f1, r0, r1;
  tfry(0u, 123u, 0u, (unsigned)layer, f0, f1);   // fold_in(key(123), layer)
  tfry(f0, f1, 0u, (unsigned)j, r0, r1);
  float g0 = -logf(-logf(u_clamped(r0)));
  float g1 = -logf(-logf(u_clamped(r1)));
  float sarg = ((l1 + g1) - (l0 + g0)) * (1.0f / TAU_C);
  float soft = 1.f / (1.f + expf(-sarg));
  int o = idxT[j];
  rows[E_EDGES + j] = (int)((float)cand[o] * soft);
  cols[E_EDGES + j] = (int)((float)cand[N_CAND + o] * soft);
}

// ---------------- adjacency assembly + normalization ----------------

__global__ void k_baseedges(const int* __restrict__ ei, int* __restrict__ rows,
                            int* __restrict__ cols) {
  int i = blockIdx.x * blockDim.x + threadIdx.x;
  if (i < E_EDGES) {
    rows[i] = ei[i];
    cols[i] = ei[E_EDGES + i];
  }
}

__global__ void k_diag(int* __restrict__ rows, int* __restrict__ cols) {
  int i = blockIdx.x * blockDim.x + threadIdx.x;
  if (i < N_NODES) {
    rows[E_EDGES + N_CAND + i] = i;
    cols[E_EDGES + N_CAND + i] = i;
  }
}

__global__ void k_zero(float* __restrict__ p, int n) {
  int i = blockIdx.x * blockDim.x + threadIdx.x;
  if (i < n) p[i] = 0.f;
}

__global__ void k_degacc(const int* __restrict__ rows, float* __restrict__ deg) {
  int i = blockIdx.x * blockDim.x + threadIdx.x;
  if (i < NNZ_TOT) atomicAdd(&deg[rows[i]], 1.0f);
}

__global__ void k_dinv(float* __restrict__ deg) {
  int i = blockIdx.x * blockDim.x + threadIdx.x;
  if (i < N_NODES) deg[i] = 1.0f / sqrtf(deg[i] + EPS_C);
}

__global__ void k_nv(const int* __restrict__ rows, const int* __restrict__ cols,
                     const float* __restrict__ dinv, float* __restrict__ nv,
                     float scale) {
  int i = blockIdx.x * blockDim.x + threadIdx.x;
  if (i < NNZ_TOT) nv[i] = dinv[rows[i]] * dinv[cols[i]] * scale;
}

// ---------------- fused dual SpMM scatter: out[c] += nv * in[r]  (128 dims) ----------------

__global__ __launch_bounds__(256) void k_scatter(
    const int* __restrict__ rows, const int* __restrict__ cols,
    const float* __restrict__ nv, const float* __restrict__ engIn,
    const float* __restrict__ hidIn, float* __restrict__ engOut,
    float* __restrict__ hidOut) {
  int wid = (int)((blockIdx.x * blockDim.x + threadIdx.x) >> 5);
  int lane = threadIdx.x & 31;
  if (wid >= NNZ_TOT) return;
  int r = rows[wid], c = cols[wid];
  float w = nv[wid];
  const float* er = engIn + (size_t)r * H_DIM;
  const float* hr = hidIn + (size_t)r * H_DIM;
  float* eo = engOut + (size_t)c * H_DIM;
  float* ho = hidOut + (size_t)c * H_DIM;
#pragma unroll
  for (int d = 0; d < 4; ++d) {
    int col = lane * 4 + d;
    atomicAdd(&eo[col], w * er[col]);
    atomicAdd(&ho[col], w * hr[col]);
  }
}

// ---------------- energies = logsumexp(energy_input @ W1 + b1 + EPS) ----------------

__global__ __launch_bounds__(256) void k_energy(const float* __restrict__ ein,
                                                const float* __restrict__ W1,
                                                const float* __restrict__ b1,
                                                float* __restrict__ energies) {
  __shared__ float sm[16][H_DIM];
  __shared__ float red[16][16];
  const int tid = threadIdx.x;
  const int row0 = blockIdx.x * 16;
  for (int t = tid; t < 16 * H_DIM; t += 256)
    sm[t >> 7][t & 127] = ein[(size_t)row0 * H_DIM + t];
  __syncthreads();
  const int r = tid >> 4, c = tid & 15;
  float y = b1[c] + EPS_C;
  for (int k = 0; k < H_DIM; ++k) y += sm[r][k] * W1[k * C_DIM + c];
  red[r][c] = y;
  __syncthreads();
  if (c == 0) {
    float mx = red[r][0];
    for (int t = 1; t < 16; ++t) mx = fmaxf(mx, red[r][t]);
    float se = 0.f;
    for (int t = 0; t < 16; ++t) se += expf(red[r][t] - mx);
    energies[row0 + r] = mx + logf(se);
  }
}

// ---------------- hidden = alpha*LN(relu(spmm)) + (1-alpha)*inp   (in-place safe) ----------

__global__ __launch_bounds__(256) void k_hidup(float* __restrict__ tmpH,
                                               const float* __restrict__ inp,
                                               const float* __restrict__ g,
                                               const float* __restrict__ bb) {
  __shared__ float sv[16][H_DIM];
  __shared__ float rs[16][16], rq[16][16];
  __shared__ float rm[16], rr[16];
  const int tid = threadIdx.x;
  const int row = tid >> 4, sub = tid & 15;
  const int gr = blockIdx.x * 16 + row;
  float s = 0.f, q = 0.f;
#pragma unroll
  for (int d = 0; d < 8; ++d) {
    int col = sub * 8 + d;
    float v = fmaxf(tmpH[(size_t)gr * H_DIM + col], 0.f);
    sv[row][col] = v;
    s += v;
    q += v * v;
  }
  rs[row][sub] = s;
  rq[row][sub] = q;
  __syncthreads();
  if (sub == 0) {
    float S = 0.f, Q = 0.f;
    for (int t = 0; t < 16; ++t) { S += rs[row][t]; Q += rq[row][t]; }
    float m = S * (1.0f / H_DIM);
    float var = Q * (1.0f / H_DIM) - m * m;
    rm[row] = m;
    rr[row] = rsqrtf(var + 1e-5f);
  }
  __syncthreads();
#pragma unroll
  for (int d = 0; d < 8; ++d) {
    int col = sub * 8 + d;
    float y = (sv[row][col] - rm[row]) * rr[row] * g[col] + bb[col];
    tmpH[(size_t)gr * H_DIM + col] =
        ALPHA_C * y + (1.0f - ALPHA_C) * inp[(size_t)gr * H_DIM + col];
  }
}

// ---------------- final: out = hidden @ W1 + b1 via V_WMMA_F32_16X16X4_F32 ----------------

__global__ __launch_bounds__(32) void k_outgemm(const float* __restrict__ hid,
                                                const float* __restrict__ W1,
                                                const float* __restrict__ b1,
                                                float* __restrict__ out) {
  const int lane = threadIdx.x;
  const int row0 = blockIdx.x * 16;
  const int hl = lane & 15;
  const int hi = lane >> 4;
  v8f acc = {};
  for (int k = 0; k < H_DIM; k += 4) {
    v2f a, b;
    const float* ar = hid + (size_t)(row0 + hl) * H_DIM + k + hi * 2;
    a[0] = ar[0];
    a[1] = ar[1];
    const float* bk = W1 + (size_t)(k + hi * 2) * C_DIM + hl;
    b[0] = bk[0];
    b[1] = bk[C_DIM];
    acc = __builtin_amdgcn_wmma_f32_16x16x4_f32(false, a, false, b, (short)0, acc,
                                                false, false);
  }
  const float bias = b1[hl];
#pragma unroll
  for (int r = 0; r < 8; ++r)
    out[(size_t)(row0 + r + hi * 8) * C_DIM + hl] = acc[r] + bias;
}

// ---------------- host launcher ----------------

extern "C" void kernel_launch(void* const* d_in, const int* in_sizes, int n_in,
                              void* d_out, int out_size, void* d_ws, size_t ws_size,
                              hipStream_t stream) {
  (void)in_sizes; (void)n_in; (void)out_size; (void)ws_size;
  const float* feat = (const float*)d_in[0];
  const float* W0 = (const float*)d_in[1];
  const float* b0 = (const float*)d_in[2];
  const float* W1 = (const float*)d_in[3];
  const float* b1 = (const float*)d_in[4];
  const float* ln_g = (const float*)d_in[5];
  const float* ln_b = (const float*)d_in[6];
  const int* ei = (const int*)d_in[7];
  const int* cand = (const int*)d_in[8];
  float* out = (float*)d_out;

  char* base = (char*)d_ws;
  size_t off = 0;
  auto alloc = [&](size_t bytes) -> char* {
    char* p = base + off;
    off = (off + bytes + 255) & ~(size_t)255;
    return p;
  };
  const size_t NH = (size_t)N_NODES * H_DIM;
  float* inp  = (float*)alloc(NH * sizeof(float));  // h after LN (also initial hidden/energy_input)
  float* hidA = (float*)alloc(NH * sizeof(float));
  float* hidB = (float*)alloc(NH * sizeof(float));
  float* engA = (float*)alloc(NH * sizeof(float));
  float* engB = (float*)alloc(NH * sizeof(float));
  float* energies = (float*)alloc(N_NODES * sizeof(float));
  float* keyT = (float*)alloc((size_t)CAND_PAD * sizeof(float));
  int* idxT = (int*)alloc((size_t)CAND_PAD * sizeof(int));
  int* rows = (int*)alloc((size_t)NNZ_TOT * sizeof(int));
  int* cols = (int*)alloc((size_t)NNZ_TOT * sizeof(int));
  float* nv = (float*)alloc((size_t)NNZ_TOT * sizeof(float));
  float* deg = (float*)alloc(N_NODES * sizeof(float));
  float* stats = (float*)alloc(64);

  // h = LN(feature@W0+b0); energies = LSE(h@W1+b1)
  k_gemm0_ln_energy<<<N_NODES / 16, 256, 0, stream>>>(feat, W0, b0, W1, b1, ln_g,
                                                      ln_b, inp, energies);
  // static parts of the edge list
  k_baseedges<<<(E_EDGES + 255) / 256, 256, 0, stream>>>(ei, rows, cols);
  k_diag<<<(N_NODES + 255) / 256, 256, 0, stream>>>(rows, cols);

  const float* hid_cur = inp;
  const float* eng_cur = inp;
  float* hid_bufs[2] = {hidA, hidB};
  float* eng_bufs[2] = {engA, engB};

  for (int layer = 0; layer < N_LAYERS; ++layer) {
    float* hid_nxt = hid_bufs[layer];
    float* eng_nxt = eng_bufs[layer];

    k_statinit<<<1, 1, 0, stream>>>(stats);
    k_score<<<N_CAND * 32 / 256, 256, 0, stream>>>(hid_cur, energies, cand, keyT,
                                                   idxT, stats);
    k_pad<<<((CAND_PAD - N_CAND) + 255) / 256, 256, 0, stream>>>(keyT, idxT);
    for (int k = 2; k <= CAND_PAD; k <<= 1)
      for (int j = k >> 1; j > 0; j >>= 1)
        k_bitonic<<<CAND_PAD / 256, 256, 0, stream>>>(keyT, idxT, j, k);
    k_sel<<<(N_CAND + 255) / 256, 256, 0, stream>>>(keyT, idxT, cand, stats, rows,
                                                    cols, layer);

    k_zero<<<(N_NODES + 255) / 256, 256, 0, stream>>>(deg, N_NODES);
    k_degacc<<<(NNZ_TOT + 255) / 256, 256, 0, stream>>>(rows, deg);
    k_dinv<<<(N_NODES + 255) / 256, 256, 0, stream>>>(deg);
    float scale = (float)(N_LAYERS - layer) / (float)N_LAYERS;
    k_nv<<<(NNZ_TOT + 255) / 256, 256, 0, stream>>>(rows, cols, deg, nv, scale);

    k_zero<<<(int)((NH + 255) / 256), 256, 0, stream>>>(eng_nxt, (int)NH);
    k_zero<<<(int)((NH + 255) / 256), 256, 0, stream>>>(hid_nxt, (int)NH);
    k_scatter<<<(int)(((size_t)NNZ_TOT * 32 + 255) / 256), 256, 0, stream>>>(
        rows, cols, nv, eng_cur, hid_cur, eng_nxt, hid_nxt);

    k_energy<<<N_NODES / 16, 256, 0, stream>>>(eng_nxt, W1, b1, energies);
    k_hidup<<<N_NODES / 16, 256, 0, stream>>>(hid_nxt, inp,
                                              ln_g + (size_t)(layer + 1) * H_DIM,
                                              ln_b + (size_t)(layer + 1) * H_DIM);
    hid_cur = hid_nxt;
    eng_cur = eng_nxt;
  }

  k_outgemm<<<N_NODES / 16, 32, 0, stream>>>(hid_cur, W1, b1, out);
}